// DistMultPredictor_11716670783784
// MI455X (gfx1250) — compile-verified
//
#include <hip/hip_runtime.h>

typedef float v2f __attribute__((ext_vector_type(2)));
typedef float v8f __attribute__((ext_vector_type(8)));

#define N_HID 64

// DistMult edge scoring, 16 edges per wave32 via V_WMMA_F32_16X16X4_F32.
// score(16 edges) = diag( (h_u .* W_r)[16x64]  x  h_v^T[64x16] ),
// accumulated as 16 K-chunks of 4 dims in full f32 precision.
__global__ __launch_bounds__(256) void distmult_wmma_kernel(
    const float* __restrict__ h,
    const float* __restrict__ W,
    const int*   __restrict__ src_idx,
    const int*   __restrict__ dst_idx,
    const int*   __restrict__ rel_idx,
    float*       __restrict__ out,
    int n_edges)
{
    const int lane          = threadIdx.x & 31;
    const int wave          = threadIdx.x >> 5;
    const int wavesPerBlock = blockDim.x >> 5;
    const int gwave         = blockIdx.x * wavesPerBlock + wave;
    const int nwaves        = gridDim.x * wavesPerBlock;

    const int half = lane >> 4;   // 0 -> K dims {0,1} of each chunk, 1 -> {2,3}
    const int m    = lane & 15;   // edge slot (M for A, N for B) within the group

    const int ngroups = (n_edges + 15) >> 4;

    for (int g = gwave; g < ngroups; g += nwaves) {
        const int base = g << 4;

        // Clamp tail edges (branch-free: EXEC stays all-ones through the WMMAs)
        int e = base + m;
        if (e >= n_edges) e = n_edges - 1;

        const long long si = (long long)src_idx[e];
        const long long di = (long long)dst_idx[e];
        const long long ri = (long long)rel_idx[e];

        const float* __restrict__ up = h + si * N_HID;  // h_u row (256B, L2-resident table)
        const float* __restrict__ vp = h + di * N_HID;  // h_v row
        const float* __restrict__ wp = W + ri * N_HID;  // relation row (tiny, L0-hot)

        v8f acc = {0.f, 0.f, 0.f, 0.f, 0.f, 0.f, 0.f, 0.f};

        const int lsel = half << 1;  // per-lane K sub-offset within each 4-dim chunk

        #pragma unroll
        for (int k = 0; k < 16; ++k) {
            const int koff = (k << 2) + lsel;
            // A-matrix fragment: (h_u .* W_r) chunk, 2 f32 per lane
            v2f a2 = *(const v2f*)(up + koff);
            v2f w2 = *(const v2f*)(wp + koff);
            // B-matrix fragment: h_v chunk (4x16, mirrored layout -> same addressing)
            v2f b2 = *(const v2f*)(vp + koff);
            v2f A;
            A.x = a2.x * w2.x;
            A.y = a2.y * w2.y;
            // D = A(16x4) x B(4x16) + C, f32 accumulate
            acc = __builtin_amdgcn_wmma_f32_16x16x4_f32(
                /*neg_a=*/false, A, /*neg_b=*/false, b2,
                /*c_mod=*/(short)0, acc, /*reuse_a=*/false, /*reuse_b=*/false);
        }

        // Diagonal extraction from the 16x16 f32 C/D layout:
        //  (m,m) for m<8  -> VGPR m,   lane m        (lanes 0..7)
        //  (m,m) for m>=8 -> VGPR m-8, lane m+16     (lanes 24..31)
        const int rr = lane & 7;
        float sc = acc[0];
        sc = (rr == 1) ? acc[1] : sc;
        sc = (rr == 2) ? acc[2] : sc;
        sc = (rr == 3) ? acc[3] : sc;
        sc = (rr == 4) ? acc[4] : sc;
        sc = (rr == 5) ? acc[5] : sc;
        sc = (rr == 6) ? acc[6] : sc;
        sc = (rr == 7) ? acc[7] : sc;

        const bool holds_score = (lane < 8) || (lane >= 24);
        const int  eo   = (lane < 8) ? lane : (lane - 16);  // edge slot 0..15
        const int  edge = base + eo;
        if (holds_score && edge < n_edges) {
            out[edge] = 1.0f / (1.0f + __expf(-sc));  // sigmoid
        }
    }
}

extern "C" void kernel_launch(void* const* d_in, const int* in_sizes, int n_in,
                              void* d_out, int out_size, void* d_ws, size_t ws_size,
                              hipStream_t stream) {
    const float* h   = (const float*)d_in[0];
    const float* W   = (const float*)d_in[1];
    const int*   src = (const int*)d_in[2];
    const int*   dst = (const int*)d_in[3];
    const int*   rel = (const int*)d_in[4];
    float*       out = (float*)d_out;

    const int n_edges = in_sizes[2];
    const int ngroups = (n_edges + 15) >> 4;

    const int threads       = 256;           // 8 wave32 per block
    const int wavesPerBlock = threads / 32;
    int blocks = (ngroups + wavesPerBlock - 1) / wavesPerBlock;
    if (blocks < 1) blocks = 1;
    if (blocks > 65535) blocks = 65535;      // grid-stride covers the rest

    distmult_wmma_kernel<<<blocks, threads, 0, stream>>>(
        h, W, src, dst, rel, out, n_edges);
}